// SpanExtractor_61615600828576
// MI455X (gfx1250) — compile-verified
//
#include <hip/hip_runtime.h>
#include <hip/hip_bf16.h>

typedef float v2f __attribute__((ext_vector_type(2)));
typedef float v4f __attribute__((ext_vector_type(4)));
typedef float v8f __attribute__((ext_vector_type(8)));

#define B_  16
#define L_  2048
#define D_  768
#define S_  256
#define K_  (2 * D_)      // 1536
#define M_  (B_ * S_)     // 4096
#define SD_ 18            // LDS row stride in floats (bank-conflict-free)
#define KC_ 16            // K chunk per LDS stage

// ---------------------------------------------------------------------------
// Kernel 1: span max/mean pooling -> cat[M_][K_]  (cat = [max(768) | mean(768)])
// One block per span, 192 threads, 4 channels per thread (float4 coalesced).
// ---------------------------------------------------------------------------
__global__ __launch_bounds__(192) void span_pool_kernel(
    const float* __restrict__ repr,      // [B_][L_][D_]
    const int*   __restrict__ spans,     // [B_][S_][2] (start, end)
    float*       __restrict__ cat)       // [M_][K_]
{
    const int span = blockIdx.x;             // 0..M_-1
    const int b    = span >> 8;              // span / S_
    const int d    = threadIdx.x * 4;        // 0..764

    const int start = spans[span * 2 + 0];
    const int end   = spans[span * 2 + 1];
    const int n     = end - start + 1;       // width+1, always >= 1

    const float* base = repr + ((size_t)b * L_ + (size_t)start) * D_ + d;

    float mx0 = -3.0e38f, mx1 = -3.0e38f, mx2 = -3.0e38f, mx3 = -3.0e38f;
    float s0 = 0.f, s1 = 0.f, s2 = 0.f, s3 = 0.f;

    for (int t = 0; t < n; ++t) {
        v4f v = *(const v4f*)(base + (size_t)t * D_);
        mx0 = fmaxf(mx0, v[0]); s0 += v[0];
        mx1 = fmaxf(mx1, v[1]); s1 += v[1];
        mx2 = fmaxf(mx2, v[2]); s2 += v[2];
        mx3 = fmaxf(mx3, v[3]); s3 += v[3];
    }

    const float inv = 1.0f / (float)n;
    float* c = cat + (size_t)span * K_ + d;
    c[0] = mx0; c[1] = mx1; c[2] = mx2; c[3] = mx3;
    c[D_ + 0] = s0 * inv; c[D_ + 1] = s1 * inv;
    c[D_ + 2] = s2 * inv; c[D_ + 3] = s3 * inv;
}

// Store 8 consecutive floats (two v4f) into LDS as 4 x ds_store_b64.
#define STORE8(dst, x0, x1) do {                       \
    *(v2f*)&(dst)[0] = (v2f){(x0)[0], (x0)[1]};        \
    *(v2f*)&(dst)[2] = (v2f){(x0)[2], (x0)[3]};        \
    *(v2f*)&(dst)[4] = (v2f){(x1)[0], (x1)[1]};        \
    *(v2f*)&(dst)[6] = (v2f){(x1)[2], (x1)[3]};        \
} while (0)

#define WMMA_F32(a, b, c) __builtin_amdgcn_wmma_f32_16x16x4_f32( \
    false, (a), false, (b), (short)0, (c), false, false)

// ---------------------------------------------------------------------------
// Kernel 2: out[M_][D_] = cat[M_][K_] * W[D_][K_]^T + bias, fp32 WMMA.
// LDS double-buffered tiled GEMM:
//   block tile 128(M) x 128(N), K chunk 16, 256 threads = 8 waves (4M x 2N),
//   wave tile 32(M) x 64(N) -> 8 v8f accumulators.
// Fragment lane layout (V_WMMA_F32_16X16X4_F32): lanes 0-15 hold {k,k+1} of
// row (lane), lanes 16-31 hold {k+2,k+3} of row (lane-16); same for B with N.
// ---------------------------------------------------------------------------
__global__ __launch_bounds__(256) void span_gemm_kernel(
    const float* __restrict__ cat,   // [M_][K_]
    const float* __restrict__ W,     // [D_][K_]
    const float* __restrict__ bias,  // [D_]
    float*       __restrict__ out)   // [M_][D_]
{
    __shared__ float lA[2][128 * SD_];
    __shared__ float lB[2][128 * SD_];

    const int tid  = threadIdx.x;
    const int lane = tid & 31;
    const int wave = tid >> 5;       // 0..7
    const int mw   = wave & 3;       // 4 waves along M
    const int nw   = wave >> 2;      // 2 waves along N

    const int mBlock = blockIdx.y * 128;
    const int nBlock = blockIdx.x * 128;

    // --- staging: each thread moves 8 floats of A and of B per chunk ---
    const int srow  = tid >> 1;            // 0..127
    const int shalf = (tid & 1) * 8;       // 0 or 8
    const float* gA = cat + (size_t)(mBlock + srow) * K_ + shalf;
    const float* gB = W   + (size_t)(nBlock + srow) * K_ + shalf;
    const int sbase = srow * SD_ + shalf;  // even -> 8B aligned LDS stores

    // --- compute-fragment indexing ---
    const int row  = lane & 15;
    const int koff = (lane >> 4) << 1;     // lanes 0-15 -> 0, 16-31 -> 2

    v8f acc[2][4] = {};

    // Prologue: stage chunk 0 into buffer 0.
    {
        v4f a0 = *(const v4f*)(gA);
        v4f a1 = *(const v4f*)(gA + 4);
        v4f b0 = *(const v4f*)(gB);
        v4f b1 = *(const v4f*)(gB + 4);
        STORE8(&lA[0][sbase], a0, a1);
        STORE8(&lB[0][sbase], b0, b1);
    }
    __syncthreads();

    const int NC = K_ / KC_;   // 96 chunks
    for (int c = 0; c < NC; ++c) {
        const int buf = c & 1;

        // Prefetch chunk c+1 into the other buffer (overlaps with compute).
        if (c + 1 < NC) {
            const float* pA = gA + (c + 1) * KC_;
            const float* pB = gB + (c + 1) * KC_;
            v4f a0 = *(const v4f*)(pA);
            v4f a1 = *(const v4f*)(pA + 4);
            v4f b0 = *(const v4f*)(pB);
            v4f b1 = *(const v4f*)(pB + 4);
            STORE8(&lA[buf ^ 1][sbase], a0, a1);
            STORE8(&lB[buf ^ 1][sbase], b0, b1);
        }

        // Compute on buffer `buf`.
        const float* A0 = &lA[buf][(mw * 32 + row) * SD_ + koff];
        const float* A1 = A0 + 16 * SD_;
        const float* B0 = &lB[buf][(nw * 64 + row) * SD_ + koff];

        #pragma unroll
        for (int kl = 0; kl < KC_; kl += 4) {
            v2f a0 = *(const v2f*)(A0 + kl);
            v2f a1 = *(const v2f*)(A1 + kl);
            #pragma unroll
            for (int j = 0; j < 4; ++j) {
                v2f b = *(const v2f*)(B0 + j * 16 * SD_ + kl);
                acc[0][j] = WMMA_F32(a0, b, acc[0][j]);
                acc[1][j] = WMMA_F32(a1, b, acc[1][j]);
            }
        }
        __syncthreads();
    }

    // Writeback with fused bias. C/D layout: VGPR r, lanes 0-15 -> M=r,
    // N=lane; lanes 16-31 -> M=8+r, N=lane-16.
    const int col = lane & 15;
    const int mb  = (lane >> 4) * 8;
    const int m0  = mBlock + mw * 32;
    const int n0  = nBlock + nw * 64;

    #pragma unroll
    for (int j = 0; j < 4; ++j) {
        const float bv = bias[n0 + j * 16 + col];
        #pragma unroll
        for (int r = 0; r < 8; ++r) {
            out[(size_t)(m0 + mb + r)      * D_ + n0 + j * 16 + col] = acc[0][j][r] + bv;
            out[(size_t)(m0 + 16 + mb + r) * D_ + n0 + j * 16 + col] = acc[1][j][r] + bv;
        }
    }
}

// ---------------------------------------------------------------------------
extern "C" void kernel_launch(void* const* d_in, const int* in_sizes, int n_in,
                              void* d_out, int out_size, void* d_ws, size_t ws_size,
                              hipStream_t stream) {
    const float* repr  = (const float*)d_in[0];  // sentence_repr [16][2048][768]
    const int*   spans = (const int*)  d_in[1];  // entity_span_indices [16][256][2]
    const float* W     = (const float*)d_in[2];  // [768][1536]
    const float* bias  = (const float*)d_in[3];  // [768]
    float*       out   = (float*)d_out;          // [4096][768]
    float*       cat   = (float*)d_ws;           // [4096][1536] = 25 MB scratch

    // Stage 1: pooled span features.
    span_pool_kernel<<<M_, 192, 0, stream>>>(repr, spans, cat);

    // Stage 2: LDS-tiled fp32 WMMA GEMM with fused bias.
    dim3 grid(D_ / 128, M_ / 128);   // (6, 32)
    span_gemm_kernel<<<grid, 256, 0, stream>>>(cat, W, bias, out);
}